// Decomposition_88716844466597
// MI455X (gfx1250) — compile-verified
//
#include <hip/hip_runtime.h>
#include <hip/hip_bf16.h>

typedef __attribute__((ext_vector_type(2))) float v2f;
typedef __attribute__((ext_vector_type(8))) float v8f;

#define DIN 64
#define DOUT 64
#define TILE_E 128
#define XS_STRIDE 68   // 64 + 4 pad: A-frag b64 reads hit 64 distinct banks
#define W_STRIDE 65    // 64 + 1 pad for B-frag column reads

// ---- W_r = sum_b comb[r,b] * bases[b]  (R blocks, each 64x64 output) ----
__global__ void compute_W_kernel(const float* __restrict__ bases,
                                 const float* __restrict__ comb,
                                 float* __restrict__ W, int B) {
  int r = blockIdx.x;
  const float* c = comb + r * B;
  for (int idx = threadIdx.x; idx < DIN * DOUT; idx += blockDim.x) {
    float acc = 0.f;
    for (int b = 0; b < B; ++b) acc += c[b] * bases[b * DIN * DOUT + idx];
    W[r * DIN * DOUT + idx] = acc;
  }
}

// ---- per-relation histogram ----
__global__ void hist_kernel(const int* __restrict__ et, int E, int* counts) {
  int e = blockIdx.x * blockDim.x + threadIdx.x;
  if (e < E) atomicAdd(&counts[et[e]], 1);
}

// ---- exclusive scan over R (tiny) ----
__global__ void scan_kernel(const int* __restrict__ counts, int* offsets,
                            int* cursor, int R) {
  if (threadIdx.x == 0 && blockIdx.x == 0) {
    int off = 0;
    for (int r = 0; r < R; ++r) { offsets[r] = off; cursor[r] = off; off += counts[r]; }
  }
}

// ---- bucket edges by relation ----
__global__ void scatter_kernel(const int* __restrict__ src, const int* __restrict__ tgt,
                               const int* __restrict__ et, const float* __restrict__ wts,
                               int E, int* cursor, int* s_src, int* s_tgt, float* s_w) {
  int e = blockIdx.x * blockDim.x + threadIdx.x;
  if (e < E) {
    int r = et[e];
    int p = atomicAdd(&cursor[r], 1);
    s_src[p] = src[e];
    s_tgt[p] = tgt[e];
    s_w[p]   = wts[e];
  }
}

// ---- main: gather 128 rows -> LDS, WMMA f32 GEMM vs W_r, atomic-scatter to out ----
__global__ void __launch_bounds__(256)
rgcn_tile_kernel(const float* __restrict__ x,
                 const float* __restrict__ W,
                 const int* __restrict__ s_src,
                 const int* __restrict__ s_tgt,
                 const float* __restrict__ s_w,
                 const int* __restrict__ offsets,
                 const int* __restrict__ counts,
                 float* __restrict__ out) {
  int r = blockIdx.y;
  int count = counts[r];
  int tile0 = (int)blockIdx.x * TILE_E;
  if (tile0 >= count) return;                       // uniform early exit
  int start = offsets[r] + tile0;
  int nE = min(TILE_E, count - tile0);

  __shared__ float xs[TILE_E * XS_STRIDE];
  __shared__ float Wl[DIN * W_STRIDE];
  __shared__ int   tg[TILE_E];

  // stage W_r into LDS (padded stride)
  for (int idx = threadIdx.x; idx < DIN * DOUT; idx += 256) {
    int k = idx >> 6, n = idx & 63;
    Wl[k * W_STRIDE + n] = W[r * DIN * DOUT + idx];
  }

  // gather + pre-scale by edge weight: 2 threads per row, float4 loads
  int row  = threadIdx.x >> 1;
  int half = threadIdx.x & 1;
  int src = 0; float wt = 0.f;
  if (row < nE) { src = s_src[start + row]; wt = s_w[start + row]; }
  if (half == 0) tg[row] = (row < nE) ? s_tgt[start + row] : -1;
  const float4* xp = (const float4*)(x + (size_t)src * DIN) + half * 8;
  float* xd = &xs[row * XS_STRIDE + half * 32];
#pragma unroll
  for (int i = 0; i < 8; ++i) {
    float4 v = xp[i];
    xd[4 * i + 0] = v.x * wt;
    xd[4 * i + 1] = v.y * wt;
    xd[4 * i + 2] = v.z * wt;
    xd[4 * i + 3] = v.w * wt;
  }
  __syncthreads();

  // wave w owns edge rows [16w, 16w+16); sweep 4 col-tiles of 16, K in steps of 4
  int lane = threadIdx.x & 31;
  int wv   = threadIdx.x >> 5;
  int hi   = lane >> 4;    // lanes 16-31 hold K+2/K+3 of A-frag
  int lm   = lane & 15;
  const float* arow = &xs[(wv * 16 + lm) * XS_STRIDE + 2 * hi];

#pragma unroll
  for (int ct = 0; ct < 4; ++ct) {
    v8f acc = {};
    const float* bcol = &Wl[(2 * hi) * W_STRIDE + ct * 16 + lm];
#pragma unroll
    for (int ks = 0; ks < 16; ++ks) {            // K = 64 in steps of 4
      v2f a; a.x = arow[ks * 4];               a.y = arow[ks * 4 + 1];
      v2f b; b.x = bcol[(ks * 4) * W_STRIDE];  b.y = bcol[(ks * 4 + 1) * W_STRIDE];
      acc = __builtin_amdgcn_wmma_f32_16x16x4_f32(
          /*neg_a=*/false, a, /*neg_b=*/false, b,
          /*c_mod=*/(short)0, acc, /*reuse_a=*/false, /*reuse_b=*/false);
    }
    // C/D layout: vgpr j -> M = j + 8*hi, N = lane&15
#pragma unroll
    for (int j = 0; j < 8; ++j) {
      int m = wv * 16 + 8 * hi + j;
      int t = tg[m];
      if (t >= 0) atomicAdd(&out[(size_t)t * DOUT + ct * 16 + lm], acc[j]);
    }
  }
}

static inline size_t align256(size_t v) { return (v + 255) & ~(size_t)255; }

extern "C" void kernel_launch(void* const* d_in, const int* in_sizes, int n_in,
                              void* d_out, int out_size, void* d_ws, size_t ws_size,
                              hipStream_t stream) {
  const float* x       = (const float*)d_in[0];
  const int*   source  = (const int*)d_in[1];
  const int*   target  = (const int*)d_in[2];
  const int*   etype   = (const int*)d_in[3];
  const float* ew      = (const float*)d_in[4];
  const float* bases   = (const float*)d_in[5];
  const float* comb    = (const float*)d_in[6];
  float*       out     = (float*)d_out;

  const int N = in_sizes[0] / DIN;
  const int E = in_sizes[1];
  const int B = in_sizes[5] / (DIN * DOUT);
  const int R = in_sizes[6] / B;

  // workspace carve-up (~14.7 MB total)
  uint8_t* w = (uint8_t*)d_ws;
  float* Wbuf    = (float*)w;  w += align256((size_t)R * DIN * DOUT * 4);
  int*   counts  = (int*)w;    w += 256;
  int*   offsets = (int*)w;    w += 256;
  int*   cursor  = (int*)w;    w += 256;
  int*   s_src   = (int*)w;    w += align256((size_t)E * 4);
  int*   s_tgt   = (int*)w;    w += align256((size_t)E * 4);
  float* s_w     = (float*)w;

  hipMemsetAsync(counts, 0, 768, stream);                       // counts+offsets+cursor
  hipMemsetAsync(out, 0, (size_t)N * DOUT * sizeof(float), stream);

  compute_W_kernel<<<R, 256, 0, stream>>>(bases, comb, Wbuf, B);

  int eblk = (E + 255) / 256;
  hist_kernel<<<eblk, 256, 0, stream>>>(etype, E, counts);
  scan_kernel<<<1, 32, 0, stream>>>(counts, offsets, cursor, R);
  scatter_kernel<<<eblk, 256, 0, stream>>>(source, target, etype, ew, E,
                                           cursor, s_src, s_tgt, s_w);

  dim3 grid((E + TILE_E - 1) / TILE_E, R);
  rgcn_tile_kernel<<<grid, 256, 0, stream>>>(x, Wbuf, s_src, s_tgt, s_w,
                                             offsets, counts, out);
}